// LocalAttentionPairBias_2370821947580
// MI455X (gfx1250) — compile-verified
//
#include <hip/hip_runtime.h>
#include <math.h>

typedef __attribute__((ext_vector_type(2))) float v2f;
typedef __attribute__((ext_vector_type(8))) float v8f;
typedef __attribute__((ext_vector_type(4))) unsigned int v4u;
typedef __attribute__((ext_vector_type(8))) int v8i;
typedef __attribute__((ext_vector_type(4))) int v4i;

#define CA 128      // c_a
#define HEADS 4
#define DH 32
#define KNN 128     // k (neighbors)
#define CPAIR 16

#if __has_builtin(__builtin_amdgcn_tensor_load_to_lds) && \
    __has_builtin(__builtin_amdgcn_s_wait_tensorcnt)
#define HAVE_TDM 1
#endif

// ---------------------------------------------------------------------------
// TDM: DMA a contiguous 2048-float (8 KB) block Global -> LDS via the Tensor
// Data Mover (TENSOR_LOAD_TO_LDS, tracked by TENSORcnt).  D# built per ISA §8:
//   group0: [1:0]=count=1, [63:32]=lds_addr, [120:64]=global_addr, [127:126]=2
//   group1: data_size=4B, tensor_dim0=2048, tensor_dim1=1, tile_dim0=2048,
//           tile_dim1=1, tensor_dim0_stride=2048
//   groups 2/3: zero (<=2D tensor)
// This toolchain exposes the 6-arg builtin form:
//   (uint32x4 g0, int32x8 g1, int32x4, int32x4, int32x8, i32 cpol)
// ---------------------------------------------------------------------------
#ifdef HAVE_TDM
__device__ __forceinline__ void tdm_load_contig_8k(const float* gsrc, float* lds_dst)
{
    unsigned long long ga = (unsigned long long)(size_t)gsrc;
    unsigned int lds_off  = (unsigned int)(size_t)lds_dst; // flat LDS-aperture addr: low 32 bits = LDS offset

    v4u g0;
    g0.x = 1u;                                              // count=1 (valid), not restore, no gather
    g0.y = lds_off;                                         // lds_addr
    g0.z = (unsigned int)(ga & 0xFFFFFFFFull);              // global_addr[31:0]
    g0.w = (unsigned int)((ga >> 32) & 0x1FFFFFFull) | (2u << 30); // global_addr[56:32] | type=2

    v8i g1;
    g1[0] = (int)(2u << 16);                                // workgroup_mask=0, data_size=2 (4B)
    g1[1] = (int)((2048u & 0xFFFFu) << 16);                 // tensor_dim0[15:0]
    g1[2] = (int)((2048u >> 16) | (1u << 16));              // tensor_dim0[31:16], tensor_dim1[15:0]=1
    g1[3] = (int)(2048u << 16);                             // tensor_dim1[31:16]=0, tile_dim0=2048
    g1[4] = 1;                                              // tile_dim1=1, tile_dim2=0
    g1[5] = 2048;                                           // tensor_dim0_stride[31:0]
    g1[6] = 0;                                              // stride0[47:32]=0, stride1[15:0]=0
    g1[7] = 0;

    v4i z4 = {0, 0, 0, 0};
    v8i z8 = {0, 0, 0, 0, 0, 0, 0, 0};
    __builtin_amdgcn_tensor_load_to_lds(g0, g1, z4, z4, z8, 0);
}
#endif

// ---------------------------------------------------------------------------
// WMMA f32 16x16x4 tile GEMM:  C[16x16] = A[16x128] (LDS, row-major) *
//                              W^T tile  where W is row-major [co][ci] global.
// A-frag (32-bit 16x4): lane(0-15)=M, VGPR0:K=0 /VGPR1:K=1 ; lanes16-31: K=2,3
// B-frag mirrored with N in lane[3:0].
// C/D: VGPR r -> M = r + 8*(lane>>4), N = lane&15.
// ---------------------------------------------------------------------------
__device__ __forceinline__ v8f wmma_tile_f32(const float* __restrict__ a_s,
                                             const float* __restrict__ W,
                                             int nbase, int lane)
{
    v8f c = {0.f, 0.f, 0.f, 0.f, 0.f, 0.f, 0.f, 0.f};
    const int m     = lane & 15;
    const int khalf = lane >> 4;              // 0 or 1
    const float* wrow = W + (size_t)(nbase + (lane & 15)) * CA;
#pragma unroll
    for (int kb = 0; kb < CA; kb += 4) {
        v2f a = *(const v2f*)(a_s + m * CA + kb + 2 * khalf);
        v2f b = *(const v2f*)(wrow + kb + 2 * khalf);
        c = __builtin_amdgcn_wmma_f32_16x16x4_f32(
                /*neg_a=*/false, a, /*neg_b=*/false, b,
                /*c_mod=*/(short)0, c, /*reuse_a=*/false, /*reuse_b=*/false);
    }
    return c;
}

__device__ __forceinline__ void store_ctile_lds(float* __restrict__ dst,
                                                v8f c, int nbase, int lane)
{
    const int n   = lane & 15;
    const int mhi = (lane >> 4) * 8;
#pragma unroll
    for (int r = 0; r < 8; ++r)
        dst[(mhi + r) * CA + nbase + n] = c[r];
}

// ---------------------------------------------------------------------------
// Kernel A: RMSNorm + 4 projections (q,k,v,gate) for a 16-row tile.
// 256 threads = 8 waves, wave w owns output columns [16w, 16w+16).
// ---------------------------------------------------------------------------
__global__ void __launch_bounds__(256)
proj_kernel(const float* __restrict__ Q,
            const float* __restrict__ Wq, const float* __restrict__ Wk,
            const float* __restrict__ Wv, const float* __restrict__ Wg,
            const float* __restrict__ ln1, const float* __restrict__ lnq,
            const float* __restrict__ lnk,
            float* __restrict__ qn, float* __restrict__ kn,
            float* __restrict__ vv, float* __restrict__ gg)
{
    __shared__ __align__(16) float a_s[16 * CA];
    __shared__ __align__(16) float p_s[16 * CA];
    __shared__ float sc_s[16];

    const int t    = threadIdx.x;
    const int lane = t & 31;
    const int wave = t >> 5;
    const int nbase = wave * 16;
    const size_t l0 = (size_t)blockIdx.x * 16;

    // load 16x128 input tile into LDS
#ifdef HAVE_TDM
    if (wave == 0) {
        tdm_load_contig_8k(Q + l0 * CA, a_s);
        __builtin_amdgcn_s_wait_tensorcnt(0);
    }
#else
    {
        const float4* src = (const float4*)(Q + l0 * CA);
        float4* dst = (float4*)a_s;
        for (int i = t; i < (16 * CA) / 4; i += 256) dst[i] = src[i];
    }
#endif
    __syncthreads();

    // per-row RMS scale
    if (t < 16) {
        float ss = 0.f;
        for (int cc = 0; cc < CA; ++cc) { float x = a_s[t * CA + cc]; ss += x * x; }
        sc_s[t] = rsqrtf(ss * (1.f / CA) + 1e-5f);
    }
    __syncthreads();
    // normalize in place (a = rms(Q)*ln1)
    for (int i = t; i < 16 * CA; i += 256)
        a_s[i] *= sc_s[i >> 7] * ln1[i & (CA - 1)];
    __syncthreads();

    // ---- q = rms(a @ Wq.T) * lnq ----
    {
        v8f c = wmma_tile_f32(a_s, Wq, nbase, lane);
        store_ctile_lds(p_s, c, nbase, lane);
        __syncthreads();
        if (t < 16) {
            float ss = 0.f;
            for (int cc = 0; cc < CA; ++cc) { float x = p_s[t * CA + cc]; ss += x * x; }
            sc_s[t] = rsqrtf(ss * (1.f / CA) + 1e-5f);
        }
        __syncthreads();
        for (int i = t; i < 16 * CA; i += 256)
            qn[l0 * CA + i] = p_s[i] * sc_s[i >> 7] * lnq[i & (CA - 1)];
        __syncthreads();
    }
    // ---- k = rms(a @ Wk.T) * lnk ----
    {
        v8f c = wmma_tile_f32(a_s, Wk, nbase, lane);
        store_ctile_lds(p_s, c, nbase, lane);
        __syncthreads();
        if (t < 16) {
            float ss = 0.f;
            for (int cc = 0; cc < CA; ++cc) { float x = p_s[t * CA + cc]; ss += x * x; }
            sc_s[t] = rsqrtf(ss * (1.f / CA) + 1e-5f);
        }
        __syncthreads();
        for (int i = t; i < 16 * CA; i += 256)
            kn[l0 * CA + i] = p_s[i] * sc_s[i >> 7] * lnk[i & (CA - 1)];
        __syncthreads();
    }
    // ---- v = a @ Wv.T ----
    {
        v8f c = wmma_tile_f32(a_s, Wv, nbase, lane);
        store_ctile_lds(p_s, c, nbase, lane);
        __syncthreads();
        for (int i = t; i < 16 * CA; i += 256) vv[l0 * CA + i] = p_s[i];
        __syncthreads();
    }
    // ---- g = sigmoid(a @ Wg.T) ----
    {
        v8f c = wmma_tile_f32(a_s, Wg, nbase, lane);
        store_ctile_lds(p_s, c, nbase, lane);
        __syncthreads();
        for (int i = t; i < 16 * CA; i += 256) {
            float x = p_s[i];
            gg[l0 * CA + i] = 1.f / (1.f + __expf(-x));
        }
    }
}

// ---------------------------------------------------------------------------
// Kernel B: gathered attention + pair bias + softmax + gate, one block/row.
// 128 threads = 4 waves (one wave per head for the softmax reduction).
// ---------------------------------------------------------------------------
__global__ void __launch_bounds__(128)
attn_kernel(const float* __restrict__ qn, const float* __restrict__ kn,
            const float* __restrict__ vv, const float* __restrict__ gg,
            const float* __restrict__ P,  const int* __restrict__ idxg,
            const float* __restrict__ Wb, float* __restrict__ yws, int L)
{
    __shared__ float q_s[CA];
    __shared__ int   idx_s[KNN];
    __shared__ float wb_s[HEADS * CPAIR];
    __shared__ float s_s[HEADS * KNN];
    __shared__ float a_sh[HEADS * KNN];
    __shared__ float rsum_s[HEADS];

    const int t = threadIdx.x;
    const size_t l = blockIdx.x;

    q_s[t] = qn[l * CA + t];
    const int myidx = idxg[l * KNN + t];
    idx_s[t] = myidx;
    if (t < HEADS * CPAIR) wb_s[t] = Wb[t];
    // CDNA5 prefetch of the random-gather rows (global_prefetch_b8)
    __builtin_prefetch(kn + (size_t)myidx * CA, 0, 0);
    __builtin_prefetch(vv + (size_t)myidx * CA, 0, 0);
    __builtin_prefetch(P + ((size_t)l * L + myidx) * CPAIR, 0, 0);
    __syncthreads();

    // scores: thread t handles neighbor j=t for all 4 heads
    {
        const int j = t;
        const float* krow = kn + (size_t)idx_s[j] * CA;
        const float* prow = P + ((size_t)l * L + idx_s[j]) * CPAIR;
        float4 p0 = *(const float4*)(prow + 0);
        float4 p1 = *(const float4*)(prow + 4);
        float4 p2 = *(const float4*)(prow + 8);
        float4 p3 = *(const float4*)(prow + 12);
        const float inv_sqrt_dh = 0.17677669529663687f; // 1/sqrt(32)
#pragma unroll
        for (int h = 0; h < HEADS; ++h) {
            float acc = 0.f;
#pragma unroll
            for (int e = 0; e < DH; e += 4) {
                float4 kf = *(const float4*)(krow + h * DH + e);
                float4 qf = *(const float4*)(q_s + h * DH + e);
                acc += kf.x * qf.x + kf.y * qf.y + kf.z * qf.z + kf.w * qf.w;
            }
            const float* wbh = wb_s + h * CPAIR;
            float bias =
                p0.x * wbh[0]  + p0.y * wbh[1]  + p0.z * wbh[2]  + p0.w * wbh[3] +
                p1.x * wbh[4]  + p1.y * wbh[5]  + p1.z * wbh[6]  + p1.w * wbh[7] +
                p2.x * wbh[8]  + p2.y * wbh[9]  + p2.z * wbh[10] + p2.w * wbh[11] +
                p3.x * wbh[12] + p3.y * wbh[13] + p3.z * wbh[14] + p3.w * wbh[15];
            s_s[h * KNN + j] = acc * inv_sqrt_dh + bias;
        }
    }
    __syncthreads();

    // softmax over k=128, one head per wave32
    {
        const int h = t >> 5;
        const int ln = t & 31;
        float vals[4];
        float mx = -1e30f;
#pragma unroll
        for (int r = 0; r < 4; ++r) {
            vals[r] = s_s[h * KNN + ln + 32 * r];
            mx = fmaxf(mx, vals[r]);
        }
#pragma unroll
        for (int off = 16; off > 0; off >>= 1)
            mx = fmaxf(mx, __shfl_xor(mx, off, 32));
        float ssum = 0.f;
#pragma unroll
        for (int r = 0; r < 4; ++r) {
            float p = __expf(vals[r] - mx);
            a_sh[h * KNN + ln + 32 * r] = p;
            ssum += p;
        }
#pragma unroll
        for (int off = 16; off > 0; off >>= 1)
            ssum += __shfl_xor(ssum, off, 32);
        if (ln == 0) rsum_s[h] = 1.f / ssum;
    }
    __syncthreads();

    // out[c] = sum_j attn[h(c)][j] * V[idx_j][c];  gate; store y
    {
        const int c = t;
        const int h = c >> 5;
        float acc = 0.f;
#pragma unroll 4
        for (int j = 0; j < KNN; ++j)
            acc += a_sh[h * KNN + j] * vv[(size_t)idx_s[j] * CA + c];
        yws[l * CA + c] = acc * rsum_s[h] * gg[l * CA + c];
    }
}

// ---------------------------------------------------------------------------
// Kernel C: final projection out = y @ Wo.T (WMMA, same tiling as kernel A)
// ---------------------------------------------------------------------------
__global__ void __launch_bounds__(256)
outproj_kernel(const float* __restrict__ y, const float* __restrict__ Wo,
               float* __restrict__ out)
{
    __shared__ __align__(16) float a_s[16 * CA];
    __shared__ __align__(16) float p_s[16 * CA];
    const int t    = threadIdx.x;
    const int lane = t & 31;
    const int wave = t >> 5;
    const size_t l0 = (size_t)blockIdx.x * 16;

#ifdef HAVE_TDM
    if (wave == 0) {
        tdm_load_contig_8k(y + l0 * CA, a_s);
        __builtin_amdgcn_s_wait_tensorcnt(0);
    }
#else
    {
        const float4* src = (const float4*)(y + l0 * CA);
        float4* dst = (float4*)a_s;
        for (int i = t; i < (16 * CA) / 4; i += 256) dst[i] = src[i];
    }
#endif
    __syncthreads();
    v8f c = wmma_tile_f32(a_s, Wo, wave * 16, lane);
    store_ctile_lds(p_s, c, wave * 16, lane);
    __syncthreads();
    for (int i = t; i < 16 * CA; i += 256) out[l0 * CA + i] = p_s[i];
}

// ---------------------------------------------------------------------------
extern "C" void kernel_launch(void* const* d_in, const int* in_sizes, int n_in,
                              void* d_out, int out_size, void* d_ws, size_t ws_size,
                              hipStream_t stream)
{
    const float* Q   = (const float*)d_in[0];   // [1,L,128]
    const float* P   = (const float*)d_in[1];   // [1,L,L,16]
    const int*   idx = (const int*)d_in[2];     // [1,L,128]
    const float* Wq  = (const float*)d_in[3];
    const float* Wk  = (const float*)d_in[4];
    const float* Wv  = (const float*)d_in[5];
    const float* Wg  = (const float*)d_in[6];
    const float* Wb  = (const float*)d_in[7];   // [4,16]
    const float* Wo  = (const float*)d_in[8];
    const float* ln1 = (const float*)d_in[9];
    const float* lnq = (const float*)d_in[10];
    const float* lnk = (const float*)d_in[11];

    const int L = in_sizes[0] / CA;             // 4096

    float* qn = (float*)d_ws;
    float* kn = qn + (size_t)L * CA;
    float* vv = kn + (size_t)L * CA;
    float* gg = vv + (size_t)L * CA;
    float* yw = gg + (size_t)L * CA;

    proj_kernel<<<L / 16, 256, 0, stream>>>(Q, Wq, Wk, Wv, Wg, ln1, lnq, lnk,
                                            qn, kn, vv, gg);
    attn_kernel<<<L, 128, 0, stream>>>(qn, kn, vv, gg, P, idx, Wb, yw, L);
    outproj_kernel<<<L / 16, 256, 0, stream>>>(yw, Wo, (float*)d_out);
}